// GAMR_76751065580182
// MI455X (gfx1250) — compile-verified
//
#include <hip/hip_runtime.h>
#include <hip/hip_bf16.h>
#include <math.h>

typedef __attribute__((ext_vector_type(2))) float v2f;
typedef __attribute__((ext_vector_type(8))) float v8f;

#define BB   1024   // 4 images * 256 batch
#define NTOK 512
#define DZ   128
#define HID  512
#define G4   2048   // 4*H
#define KCAT 640    // D + H

// ---------------- workspace layout (float offsets) ----------------
#define OFF_ACOEF   0u
#define OFF_CCOEF   (OFF_ACOEF + BB*DZ)
#define OFF_ZSUM    (OFF_CCOEF + BB*DZ)
#define OFF_ZNMEAN  (OFF_ZSUM + BB*DZ)
#define OFF_ZNVAR   (OFF_ZNMEAN + BB*NTOK)
#define OFF_H       (OFF_ZNVAR + BB*NTOK)
#define OFF_C       (OFF_H + BB*HID)
#define OFF_KEYR    (OFF_C + BB*HID)
#define OFF_S       (OFF_KEYR + BB*DZ)
#define OFF_X       (OFF_S + BB*DZ)
#define OFF_GATES   (OFF_X + BB*KCAT)
#define OFF_QG      (OFF_GATES + BB*G4)
#define OFF_MV      (OFF_QG + BB*256)
#define OFF_ALLG    (OFF_MV + BB*4*DZ)
#define OFF_WCAT    (OFF_ALLG + BB*256)
#define OFF_WQG     (OFF_WCAT + G4*KCAT)
#define OFF_BGATE   (OFF_WQG + 256*HID)
#define OFF_BQG     (OFF_BGATE + G4)
#define WS_FLOATS   (OFF_BQG + 256)

static __device__ __forceinline__ float sigf(float x) { return 1.f / (1.f + expf(-x)); }

// ---------------- weight packing ----------------
__global__ void pack_weights(const float* __restrict__ W_ih, const float* __restrict__ W_hh,
                             const float* __restrict__ b_ih, const float* __restrict__ b_hh,
                             const float* __restrict__ W_q,  const float* __restrict__ b_q,
                             const float* __restrict__ W_g,  const float* __restrict__ b_g,
                             float* __restrict__ Wcat, float* __restrict__ Wqg,
                             float* __restrict__ bgate, float* __restrict__ bqg) {
  int idx = blockIdx.x * blockDim.x + threadIdx.x;
  if (idx < G4 * KCAT) {
    int j = idx / KCAT, k = idx % KCAT;
    Wcat[idx] = (k < DZ) ? W_ih[j * DZ + k] : W_hh[j * HID + (k - DZ)];
  }
  int i2 = idx - G4 * KCAT;
  if (i2 >= 0 && i2 < 256 * HID) {
    int j = i2 / HID, k = i2 % HID;
    Wqg[i2] = (j < 128) ? W_q[j * HID + k] : W_g[(j - 128) * HID + k];
  }
  int i3 = idx - (G4 * KCAT + 256 * HID);
  if (i3 >= 0 && i3 < G4) bgate[i3] = b_ih[i3] + b_hh[i3];
  int i4 = idx - (G4 * KCAT + 256 * HID + G4);
  if (i4 >= 0 && i4 < 256) bqg[i4] = (i4 < 128) ? b_q[i4] : b_g[i4 - 128];
}

__global__ void zero_f(float* __restrict__ p, int n) {
  int i = blockIdx.x * blockDim.x + threadIdx.x;
  if (i < n) p[i] = 0.f;
}

// ---------------- context norm stats: A,C affine coefs + Zsum ----------------
__global__ void __launch_bounds__(128)
ctx_stats(const float* __restrict__ z, const float* __restrict__ gamma1,
          const float* __restrict__ beta1, float* __restrict__ Acoef,
          float* __restrict__ Ccoef, float* __restrict__ Zsum) {
  int bb = blockIdx.x, d = threadIdx.x;
  const float* zp = z + (size_t)bb * (NTOK * DZ) + d;
  float s = 0.f, s2 = 0.f;
  for (int n = 0; n < NTOK; ++n) { float v = zp[(size_t)n * DZ]; s += v; s2 += v * v; }
  float mu  = s * (1.f / NTOK);
  float var = (s2 - (float)NTOK * mu * mu) * (1.f / (NTOK - 1));  // ddof=1
  float A   = gamma1[d] * rsqrtf(var + 1e-8f);
  float C   = beta1[d] - mu * A;
  Acoef[bb * DZ + d] = A;
  Ccoef[bb * DZ + d] = C;
  Zsum[bb * DZ + d]  = A * s + (float)NTOK * C;  // sum_n of normalized z
}

// ---------------- per-token stats of normalized z ----------------
__global__ void __launch_bounds__(256)
token_stats(const float* __restrict__ z, const float* __restrict__ Acoef,
            const float* __restrict__ Ccoef, float* __restrict__ znMean,
            float* __restrict__ znVar) {
  int bb = blockIdx.x;
  int tid = threadIdx.x, lane = tid & 31, wv = tid >> 5;
  __shared__ float As[DZ], Cs[DZ];
  if (tid < DZ) { As[tid] = Acoef[bb * DZ + tid]; Cs[tid] = Ccoef[bb * DZ + tid]; }
  __syncthreads();
  const float* zb = z + (size_t)bb * (NTOK * DZ);
  for (int n = wv; n < NTOK; n += 8) {
    const float* zr = zb + (size_t)n * DZ;
    float m1 = 0.f, m2 = 0.f;
#pragma unroll
    for (int q = 0; q < 4; ++q) {
      int d = lane + q * 32;
      float v = As[d] * zr[d] + Cs[d];
      m1 += v; m2 += v * v;
    }
    for (int off = 16; off; off >>= 1) { m1 += __shfl_xor(m1, off); m2 += __shfl_xor(m2, off); }
    if (lane == 0) {
      m1 *= (1.f / DZ); m2 *= (1.f / DZ);
      znMean[bb * NTOK + n] = m1;
      znVar[bb * NTOK + n]  = m2 - m1 * m1;   // population var
    }
  }
}

// ---------------- generic fp32 WMMA GEMM: out[M,N] = A[M,K] * W[N,K]^T + bias ----------------
// per wave: 16x64 tile; K chunked by 16 so 20 loads are in flight per 16-WMMA burst
__global__ void __launch_bounds__(256)
wmma_gemm_f32(const float* __restrict__ A, const float* __restrict__ W,
              const float* __restrict__ bias, float* __restrict__ out,
              int M, int N, int K, int reluStart) {
  const int lane = threadIdx.x & 31;
  const int wid  = blockIdx.x * (blockDim.x >> 5) + (threadIdx.x >> 5);
  const int mtiles = M >> 4, ngroups = N >> 6;
  if (wid >= mtiles * ngroups) return;
  const int mtile = wid % mtiles;
  const int ng    = wid / mtiles;
  const int lh = lane & 15, hi = lane >> 4, koff = hi << 1;
  const int rowA = (mtile << 4) + lh;
  const float* aptr = A + (size_t)rowA * K + koff;
  const v8f zero8 = {0.f, 0.f, 0.f, 0.f, 0.f, 0.f, 0.f, 0.f};
  v8f acc[4];
  int ncol[4];
  const float* bptr[4];
#pragma unroll
  for (int j = 0; j < 4; ++j) {
    ncol[j] = (ng << 6) + (j << 4) + lh;
    bptr[j] = W + (size_t)ncol[j] * K + koff;
    acc[j]  = zero8;
  }
  for (int k0 = 0; k0 < K; k0 += 16) {   // K is a multiple of 16 for all our GEMMs
    v2f a[4];
    v2f b[4][4];
#pragma unroll
    for (int s = 0; s < 4; ++s) a[s] = *(const v2f*)(aptr + k0 + 4 * s);
#pragma unroll
    for (int j = 0; j < 4; ++j)
#pragma unroll
      for (int s = 0; s < 4; ++s) b[j][s] = *(const v2f*)(bptr[j] + k0 + 4 * s);
#pragma unroll
    for (int s = 0; s < 4; ++s)
#pragma unroll
      for (int j = 0; j < 4; ++j)
        acc[j] = __builtin_amdgcn_wmma_f32_16x16x4_f32(false, a[s], false, b[j][s],
                                                       (short)0, acc[j], false, false);
  }
#pragma unroll
  for (int j = 0; j < 4; ++j) {
    float bv = bias ? bias[ncol[j]] : 0.f;
    bool doRelu = (ncol[j] >= reluStart);
#pragma unroll
    for (int v = 0; v < 8; ++v) {
      int m = (mtile << 4) + v + (hi << 3);
      float val = acc[j][v] + bv;
      if (doRelu) val = fmaxf(val, 0.f);
      out[(size_t)m * N + ncol[j]] = val;
    }
  }
}

// ---------------- X = [key_r | h] ----------------
__global__ void pack_x(const float* __restrict__ keyr, const float* __restrict__ h,
                       float* __restrict__ X) {
  int idx = blockIdx.x * blockDim.x + threadIdx.x;
  if (idx >= BB * KCAT) return;
  int bb = idx / KCAT, k = idx % KCAT;
  X[idx] = (k < DZ) ? keyr[bb * DZ + k] : h[bb * HID + (k - DZ)];
}

// ---------------- LSTM cell ----------------
__global__ void lstm_cell(const float* __restrict__ gates, float* __restrict__ h,
                          float* __restrict__ c) {
  int idx = blockIdx.x * blockDim.x + threadIdx.x;
  if (idx >= BB * HID) return;
  int bb = idx >> 9, j = idx & 511;
  const float* g = gates + (size_t)bb * G4;
  float ig = g[j], fg = g[HID + j], gg = g[2 * HID + j], og = g[3 * HID + j];
  float cn = sigf(fg) * c[idx] + sigf(ig) * tanhf(gg);
  c[idx] = cn;
  h[idx] = sigf(og) * tanhf(cn);
}

// ---------------- fused attention step (one block per bb) ----------------
// z tile [512,128] f32 (256KB) is staged into LDS with async-to-LDS copies
// (ASYNCcnt), overlapped with the query-stat reductions; both z passes then
// run out of LDS instead of re-reading global memory.
__global__ void __launch_bounds__(256)
attention_step(const float* __restrict__ z, const float* __restrict__ qg,
               const float* __restrict__ Acoef, const float* __restrict__ Ccoef,
               const float* __restrict__ Zsum, const float* __restrict__ znMean,
               const float* __restrict__ znVar, const float* __restrict__ ln_g,
               const float* __restrict__ ln_b, float* __restrict__ S,
               float* __restrict__ keyr, float* __restrict__ Mv, int t) {
  int bb = blockIdx.x;
  int tid = threadIdx.x, lane = tid & 31, wv = tid >> 5;
  extern __shared__ __align__(16) float zL[];   // NTOK*DZ floats (256KB dynamic LDS)
  __shared__ float qs[DZ], aqs[DZ], As[DZ], Cs[DZ];
  __shared__ float rb[DZ];
  __shared__ float invs[NTOK];
  __shared__ float part[256];
  __shared__ float scal[8];      // 0 qmean, 1 qvar, 2 cq, 3 ssum, 4 smean
  __shared__ float sred[8], mred[8];

  const float* zb = z + (size_t)bb * (NTOK * DZ);

  // kick off async global->LDS copy of the whole z tile (16B per op, 64 ops/thread)
  for (int i = tid; i < (NTOK * DZ) / 4; i += 256) {
    const float* gp = zb + (size_t)i * 4;
    unsigned lo = (unsigned)(uintptr_t)(zL + (size_t)i * 4);
    asm volatile("global_load_async_to_lds_b128 %0, %1, off"
                 :: "v"(lo), "v"(gp) : "memory");
  }

  float qv = 0.f;
  if (tid < DZ) {
    qv = qg[bb * 256 + tid];
    float A = Acoef[bb * DZ + tid], C = Ccoef[bb * DZ + tid];
    qs[tid] = qv; As[tid] = A; Cs[tid] = C; aqs[tid] = A * qv;
  }
  __syncthreads();
  // block reductions over 128 values: qmean, qvar, cq = sum C*q
  if (tid < DZ) rb[tid] = qv;
  __syncthreads();
  for (int s = 64; s >= 1; s >>= 1) { if (tid < s) rb[tid] += rb[tid + s]; __syncthreads(); }
  if (tid == 0) scal[0] = rb[0] * (1.f / DZ);
  __syncthreads();
  if (tid < DZ) rb[tid] = qv * qv;
  __syncthreads();
  for (int s = 64; s >= 1; s >>= 1) { if (tid < s) rb[tid] += rb[tid + s]; __syncthreads(); }
  if (tid == 0) { float m2 = rb[0] * (1.f / DZ); scal[1] = m2 - scal[0] * scal[0]; }
  __syncthreads();
  if (tid < DZ) rb[tid] = Cs[tid] * qv;
  __syncthreads();
  for (int s = 64; s >= 1; s >>= 1) { if (tid < s) rb[tid] += rb[tid + s]; __syncthreads(); }
  if (tid == 0) scal[2] = rb[0];
  __syncthreads();
  float qmean = scal[0], qvar = scal[1], cq = scal[2];

  // wait for the async tile, then make it visible to all waves
  asm volatile("s_wait_asynccnt 0x0" ::: "memory");
  __syncthreads();

  // phase 1: per-token  zdot = z.q  ->  invsig;  accumulate ssum, smean  (reads LDS)
  float ssum_l = 0.f, smean_l = 0.f;
  for (int n = wv; n < NTOK; n += 8) {
    const float* zr = zL + n * DZ;
    float p = 0.f;
#pragma unroll
    for (int q = 0; q < 4; ++q) { int d = lane + q * 32; p += aqs[d] * zr[d]; }
    for (int off = 16; off; off >>= 1) p += __shfl_xor(p, off);
    float zdot = p + cq;
    float znm  = znMean[bb * NTOK + n];
    float var  = znVar[bb * NTOK + n] + qvar + 2.f * (zdot * (1.f / DZ) - znm * qmean);
    float is   = rsqrtf(var + 1e-5f);
    if (lane == 0) { invs[n] = is; ssum_l += is; smean_l += is * znm; }
  }
  if (lane == 0) { sred[wv] = ssum_l; mred[wv] = smean_l; }
  __syncthreads();
  if (tid == 0) {
    float a = 0.f, b = 0.f;
    for (int i = 0; i < 8; ++i) { a += sred[i]; b += mred[i]; }
    scal[3] = a; scal[4] = b;
  }
  __syncthreads();

  // phase 2: acc_z[d] = sum_n invs[n] * z_raw[n,d]   (reads LDS)
  int d = tid & 127, half = tid >> 7;
  float acc = 0.f;
  for (int n = half * 256; n < half * 256 + 256; ++n) acc += invs[n] * zL[n * DZ + d];
  part[tid] = acc;
  __syncthreads();
  if (tid < DZ) {
    float accz = part[tid] + part[tid + 128];
    float ssum = scal[3], smean = scal[4];
    float A = As[d], C = Cs[d];
    float sz = A * accz + C * ssum;                       // sum_n invs * z_norm[n,d]
    float wz = ln_g[d] * (sz - smean + (qs[d] - qmean) * ssum) + (float)NTOK * ln_b[d];
    float zt = wz * Zsum[bb * DZ + d];
    float Sn = S[bb * DZ + d] + zt;
    S[bb * DZ + d]    = Sn;
    keyr[bb * DZ + d] = qg[bb * 256 + 128 + d] * Sn;      // g already relu'd
    Mv[bb * 4 * DZ + t * DZ + d] = zt;
  }
}

// ---------------- fused pair MLP (one block per bb), WMMA f32, LDS staged ----------------
__global__ void __launch_bounds__(256)
pair_mlp(const float* __restrict__ Mv, const float* __restrict__ W_gh,
         const float* __restrict__ b_gh, const float* __restrict__ W_go,
         const float* __restrict__ b_go, float* __restrict__ allg) {
  int bb = blockIdx.x;
  int tid = threadIdx.x, lane = tid & 31, wv = tid >> 5;
  __shared__ __align__(16) float pairL[16 * 256];
  __shared__ __align__(16) float ghL[16 * 512];
  const float* mvb = Mv + (size_t)bb * (4 * DZ);
  for (int i = tid; i < 16 * 256; i += 256) {
    int p = i >> 8, k = i & 255;
    pairL[i] = (k < DZ) ? mvb[(p >> 2) * DZ + k] : mvb[(p & 3) * DZ + (k - DZ)];
  }
  __syncthreads();
  const int lh = lane & 15, hi = lane >> 4, koff = hi << 1;
  const v8f zero8 = {0.f, 0.f, 0.f, 0.f, 0.f, 0.f, 0.f, 0.f};
  // GEMM1: gh[16,512] = relu(pair[16,256] * W_gh^T + b_gh)  -> LDS
#pragma unroll
  for (int jj = 0; jj < 4; ++jj) {
    int n = (wv * 4 + jj) * 16 + lh;
    const float* wp = W_gh + (size_t)n * 256 + koff;
    const float* ap = pairL + lh * 256 + koff;
    v8f acc = zero8;
    for (int k0 = 0; k0 < 256; k0 += 16) {
      v2f a[4], b[4];
#pragma unroll
      for (int s = 0; s < 4; ++s) a[s] = *(const v2f*)(ap + k0 + 4 * s);
#pragma unroll
      for (int s = 0; s < 4; ++s) b[s] = *(const v2f*)(wp + k0 + 4 * s);
#pragma unroll
      for (int s = 0; s < 4; ++s)
        acc = __builtin_amdgcn_wmma_f32_16x16x4_f32(false, a[s], false, b[s],
                                                    (short)0, acc, false, false);
    }
    float bv = b_gh[n];
#pragma unroll
    for (int v = 0; v < 8; ++v) ghL[(v + hi * 8) * 512 + n] = fmaxf(acc[v] + bv, 0.f);
  }
  __syncthreads();
  // GEMM2: go[16,256] = relu(gh * W_go^T + b_go); column-sum over the 16 pairs
#pragma unroll
  for (int jj = 0; jj < 2; ++jj) {
    int n = (wv * 2 + jj) * 16 + lh;
    const float* wp = W_go + (size_t)n * 512 + koff;
    const float* ap = ghL + lh * 512 + koff;
    v8f acc = zero8;
    for (int k0 = 0; k0 < 512; k0 += 16) {
      v2f a[4], b[4];
#pragma unroll
      for (int s = 0; s < 4; ++s) a[s] = *(const v2f*)(ap + k0 + 4 * s);
#pragma unroll
      for (int s = 0; s < 4; ++s) b[s] = *(const v2f*)(wp + k0 + 4 * s);
#pragma unroll
      for (int s = 0; s < 4; ++s)
        acc = __builtin_amdgcn_wmma_f32_16x16x4_f32(false, a[s], false, b[s],
                                                    (short)0, acc, false, false);
    }
    float bv = b_go[n];
    float s = 0.f;
#pragma unroll
    for (int v = 0; v < 8; ++v) s += fmaxf(acc[v] + bv, 0.f);
    s += __shfl_xor(s, 16);          // combine M halves (rows 0-7 + 8-15)
    if (hi == 0) allg[bb * 256 + n] = s;
  }
}

// ---------------- final projection: out[256,4] ----------------
__global__ void __launch_bounds__(128)
final_proj(const float* __restrict__ h, const float* __restrict__ allg,
           const float* __restrict__ W_y, const float* __restrict__ b_y,
           float* __restrict__ out) {
  int b = blockIdx.x, tid = threadIdx.x;
  __shared__ float rb[128];
  for (int y = 0; y < 4; ++y) {
    float acc = 0.f;
    for (int k = tid; k < 3072; k += 128) {
      int img = k / 768, r = k % 768;
      int bb = img * 256 + b;
      float fv = (r < HID) ? h[(size_t)bb * HID + r] : allg[(size_t)bb * 256 + (r - HID)];
      acc += fv * W_y[y * 3072 + k];
    }
    rb[tid] = acc;
    __syncthreads();
    for (int s = 64; s >= 1; s >>= 1) { if (tid < s) rb[tid] += rb[tid + s]; __syncthreads(); }
    if (tid == 0) out[b * 4 + y] = rb[0] + b_y[y];
    __syncthreads();
  }
}

extern "C" void kernel_launch(void* const* d_in, const int* in_sizes, int n_in,
                              void* d_out, int out_size, void* d_ws, size_t ws_size,
                              hipStream_t stream) {
  (void)in_sizes; (void)n_in; (void)out_size; (void)ws_size;
  const float* z_seq  = (const float*)d_in[0];
  const float* W_ih   = (const float*)d_in[1];
  const float* W_hh   = (const float*)d_in[2];
  const float* b_ih   = (const float*)d_in[3];
  const float* b_hh   = (const float*)d_in[4];
  const float* W_q    = (const float*)d_in[5];
  const float* b_q    = (const float*)d_in[6];
  const float* W_g    = (const float*)d_in[7];
  const float* b_g    = (const float*)d_in[8];
  const float* ln_g   = (const float*)d_in[9];
  const float* ln_b   = (const float*)d_in[10];
  const float* gamma1 = (const float*)d_in[11];
  const float* beta1  = (const float*)d_in[12];
  const float* W_gh   = (const float*)d_in[13];
  const float* b_gh   = (const float*)d_in[14];
  const float* W_go   = (const float*)d_in[15];
  const float* b_go   = (const float*)d_in[16];
  const float* W_y    = (const float*)d_in[17];
  const float* b_y    = (const float*)d_in[18];
  float* out = (float*)d_out;
  float* ws  = (float*)d_ws;

  float* Acoef  = ws + OFF_ACOEF;
  float* Ccoef  = ws + OFF_CCOEF;
  float* Zsum   = ws + OFF_ZSUM;
  float* znMean = ws + OFF_ZNMEAN;
  float* znVar  = ws + OFF_ZNVAR;
  float* h      = ws + OFF_H;
  float* c      = ws + OFF_C;
  float* keyr   = ws + OFF_KEYR;
  float* S      = ws + OFF_S;
  float* X      = ws + OFF_X;
  float* gates  = ws + OFF_GATES;
  float* qg     = ws + OFF_QG;
  float* Mv     = ws + OFF_MV;
  float* allg   = ws + OFF_ALLG;
  float* Wcat   = ws + OFF_WCAT;
  float* Wqg    = ws + OFF_WQG;
  float* bgate  = ws + OFF_BGATE;
  float* bqg    = ws + OFF_BQG;

  // prepack weights / biases
  {
    int total = G4 * KCAT + 256 * HID + G4 + 256;
    pack_weights<<<(total + 255) / 256, 256, 0, stream>>>(W_ih, W_hh, b_ih, b_hh,
                                                          W_q, b_q, W_g, b_g,
                                                          Wcat, Wqg, bgate, bqg);
  }
  // zero recurrent state (h, c, keyr, S are contiguous)
  {
    int n = BB * (HID + HID + DZ + DZ);
    zero_f<<<(n + 255) / 256, 256, 0, stream>>>(h, n);
  }
  ctx_stats<<<BB, 128, 0, stream>>>(z_seq, gamma1, beta1, Acoef, Ccoef, Zsum);
  token_stats<<<BB, 256, 0, stream>>>(z_seq, Acoef, Ccoef, znMean, znVar);

  const size_t zTileBytes = (size_t)NTOK * DZ * sizeof(float);  // 256 KB dynamic LDS

  for (int t = 0; t < 4; ++t) {
    pack_x<<<(BB * KCAT + 255) / 256, 256, 0, stream>>>(keyr, h, X);
    // gates[BB,2048] = X[BB,640] * Wcat^T + bgate   (no relu)
    {
      int waves = (BB / 16) * (G4 / 64);            // 2048
      wmma_gemm_f32<<<waves / 8, 256, 0, stream>>>(X, Wcat, bgate, gates,
                                                   BB, G4, KCAT, G4);
    }
    lstm_cell<<<(BB * HID + 255) / 256, 256, 0, stream>>>(gates, h, c);
    // qg[BB,256] = h * Wqg^T + bqg   (relu on cols >= 128 : the g-gate)
    {
      int waves = (BB / 16) * (256 / 64);           // 256
      wmma_gemm_f32<<<waves / 8, 256, 0, stream>>>(h, Wqg, bqg, qg,
                                                   BB, 256, HID, 128);
    }
    attention_step<<<BB, 256, zTileBytes, stream>>>(z_seq, qg, Acoef, Ccoef, Zsum,
                                                    znMean, znVar, ln_g, ln_b,
                                                    S, keyr, Mv, t);
  }
  pair_mlp<<<BB, 256, 0, stream>>>(Mv, W_gh, b_gh, W_go, b_go, allg);
  final_proj<<<256, 128, 0, stream>>>(h, allg, W_y, b_y, out);
}